// GraphTransformerBlock_14912126452017
// MI455X (gfx1250) — compile-verified
//
#include <hip/hip_runtime.h>
#include <stdint.h>

typedef __attribute__((ext_vector_type(2))) float v2f;
typedef __attribute__((ext_vector_type(8))) float v8f;
typedef __attribute__((ext_vector_type(4))) int   v4i;

#define N_NODES 10000
#define N_EDGES 160000
#define DIM     128
#define HEADS   8
#define CH      128
#define FF_DIM  512

#if defined(__AMDGCN__) && __has_builtin(__builtin_amdgcn_global_load_async_to_lds_b128) && __has_builtin(__builtin_amdgcn_s_wait_asynccnt)
#define USE_ASYNC_LDS 1
typedef __attribute__((address_space(1))) v4i g_v4i;   // global (device) int4
typedef __attribute__((address_space(3))) v4i l_v4i;   // LDS (shared) int4
#else
#define USE_ASYNC_LDS 0
#endif

__device__ __forceinline__ float wave_sum32(float v) {
#pragma unroll
  for (int off = 16; off >= 1; off >>= 1) v += __shfl_xor(v, off, 32);
  return v;
}

// ---------------------------------------------------------------------------
// WMMA fp32 GEMM: out[M,NC] = A[M,K] @ W[K,NC] + bias (+optional ReLU)
// K, NC compile-time -> fully unrolled inner loop, immediate-offset loads.
// Block = 256 threads (8 waves). Block tile: 16 rows x 128 cols; each wave
// owns one 16x16 tile, accumulated with V_WMMA_F32_16X16X4_F32 over two
// independent chains for ILP. A-tile staged in LDS (async copy if available).
// ---------------------------------------------------------------------------
template <int K, int NC, int RELU>
__global__ __launch_bounds__(256) void gemm_wmma_f32(
    const float* __restrict__ A, const float* __restrict__ W,
    const float* __restrict__ bias, float* __restrict__ out) {
  constexpr int KP = K + 4;                  // padded row to dodge bank conflicts
  __shared__ __align__(16) float As[16 * KP];

  const int tid   = threadIdx.x;
  const int mtile = blockIdx.x;
  const float* Ab = A + (size_t)mtile * 16 * K;

  constexpr int C4 = K / 4;                  // float4 chunks per row
#if USE_ASYNC_LDS
  for (int i = tid; i < 16 * C4; i += 256) {
    const int r = i / C4, c = i - r * C4;
    __builtin_amdgcn_global_load_async_to_lds_b128(
        (g_v4i*)(Ab + r * K + c * 4),
        (l_v4i*)(&As[r * KP + c * 4]),
        0, 0);
  }
  __builtin_amdgcn_s_wait_asynccnt(0);
  __syncthreads();
#else
  for (int i = tid; i < 16 * C4; i += 256) {
    const int r = i / C4, c = i - r * C4;
    *(float4*)&As[r * KP + c * 4] = *(const float4*)(Ab + r * K + c * 4);
  }
  __syncthreads();
#endif

  const int wave = tid >> 5;
  const int lane = tid & 31;
  const int hh   = lane >> 4;                // half-wave 0/1
  const int lo   = lane & 15;
  const int col  = blockIdx.y * 128 + wave * 16 + lo;

  // Per-lane bases; all per-iteration offsets are compile-time immediates.
  const float* __restrict__ Wc = W + col + 2 * hh * NC;  // rows kk+2hh, kk+2hh+1
  const float* __restrict__ Ar = &As[lo * KP + 2 * hh];

  v8f acc0 = {}, acc1 = {};
#pragma unroll
  for (int kk = 0; kk < K; kk += 8) {
    {
      v2f a, b;
      a[0] = Ar[kk + 0];
      a[1] = Ar[kk + 1];
      b[0] = Wc[(kk + 0) * NC];
      b[1] = Wc[(kk + 1) * NC];
      acc0 = __builtin_amdgcn_wmma_f32_16x16x4_f32(false, a, false, b,
                                                   (short)0, acc0, false, false);
    }
    {
      v2f a, b;
      a[0] = Ar[kk + 4];
      a[1] = Ar[kk + 5];
      b[0] = Wc[(kk + 4) * NC];
      b[1] = Wc[(kk + 5) * NC];
      acc1 = __builtin_amdgcn_wmma_f32_16x16x4_f32(false, a, false, b,
                                                   (short)0, acc1, false, false);
    }
  }

  const float bb = bias[col];
#pragma unroll
  for (int r = 0; r < 8; ++r) {
    const int m = mtile * 16 + r + 8 * hh;   // C/D layout: lanes 16-31 hold M=8..15
    float val = acc0[r] + acc1[r] + bb;
    if (RELU) val = fmaxf(val, 0.0f);
    out[(size_t)m * NC + col] = val;
  }
}

// ---------------------------------------------------------------------------
// Per-(edge,head) attention logit: alpha = (q[dst,h] . k[src,h]) / sqrt(C)
// One wave per (edge,head); float4 loads, wave reduction; ordered-uint
// atomicMax for the segment max.
// ---------------------------------------------------------------------------
__global__ void edge_logits_max(const float* __restrict__ q,
                                const float* __restrict__ k,
                                const long long* __restrict__ ei,
                                float* __restrict__ alpha,
                                unsigned int* __restrict__ m_enc) {
  const int lane = threadIdx.x & 31;
  const int idx  = blockIdx.x * 8 + (threadIdx.x >> 5);   // over E*H
  const int e    = idx >> 3;
  const int h    = idx & 7;
  const int src  = (int)ei[e];
  const int dst  = (int)ei[N_EDGES + e];

  const float4 q4 = *(const float4*)&q[(size_t)dst * (HEADS * CH) + h * CH + lane * 4];
  const float4 k4 = *(const float4*)&k[(size_t)src * (HEADS * CH) + h * CH + lane * 4];
  float s = q4.x * k4.x + q4.y * k4.y + q4.z * k4.z + q4.w * k4.w;
  s = wave_sum32(s);
  if (lane == 0) {
    const float a = s * 0.08838834764831845f;             // 1/sqrt(128)
    alpha[idx] = a;
    unsigned int u = __float_as_uint(a);
    u = (u & 0x80000000u) ? ~u : (u | 0x80000000u);       // order-preserving encode
    atomicMax(&m_enc[dst * HEADS + h], u);
  }
}

// ex = exp(alpha - m[dst]);  denom[dst] += ex  (one thread per (edge,head))
__global__ void edge_exp_denom(const long long* __restrict__ ei,
                               float* __restrict__ alpha,
                               const unsigned int* __restrict__ m_enc,
                               float* __restrict__ denom) {
  const long long gid = (long long)blockIdx.x * blockDim.x + threadIdx.x;
  if (gid >= (long long)N_EDGES * HEADS) return;
  const int e = (int)(gid >> 3), h = (int)(gid & 7);
  const int dst = (int)ei[N_EDGES + e];
  const unsigned int u = m_enc[dst * HEADS + h];
  const float m = (u & 0x80000000u) ? __uint_as_float(u & 0x7FFFFFFFu)
                                    : __uint_as_float(~u);
  const float ex = __expf(alpha[gid] - m);
  alpha[gid] = ex;
  atomicAdd(&denom[dst * HEADS + h], ex);
}

// Per edge: agg[dst,:] += sum_h (ex/denom) * v[src,h,:]   (one wave per edge)
__global__ void edge_aggregate(const long long* __restrict__ ei,
                               const float* __restrict__ exbuf,
                               const float* __restrict__ denom,
                               const float* __restrict__ vbuf,
                               float* __restrict__ agg) {
  const int lane = threadIdx.x & 31;
  const int e    = blockIdx.x * 8 + (threadIdx.x >> 5);
  const int src  = (int)ei[e];
  const int dst  = (int)ei[N_EDGES + e];

  float4 acc = {0.f, 0.f, 0.f, 0.f};
#pragma unroll
  for (int h = 0; h < HEADS; ++h) {
    const float ah = exbuf[(size_t)e * HEADS + h] /
                     (denom[dst * HEADS + h] + 1e-16f);
    const float4 v4 = *(const float4*)&vbuf[(size_t)src * (HEADS * CH) + h * CH + lane * 4];
    acc.x += ah * v4.x; acc.y += ah * v4.y;
    acc.z += ah * v4.z; acc.w += ah * v4.w;
  }
  float* o = &agg[(size_t)dst * DIM + lane * 4];
  atomicAdd(o + 0, acc.x); atomicAdd(o + 1, acc.y);
  atomicAdd(o + 2, acc.z); atomicAdd(o + 3, acc.w);
}

// t = p0 + scale1*p1 + (p2 ? p2 : 0); out = LN(t)*g + b  (one wave per node)
__global__ void ln_fused(const float* __restrict__ p0,
                         const float* __restrict__ p1, float scale1,
                         const float* __restrict__ p2,
                         const float* __restrict__ g,
                         const float* __restrict__ b,
                         float* __restrict__ out) {
  const int lane = threadIdx.x & 31;
  const int node = blockIdx.x * 8 + (threadIdx.x >> 5);
  const size_t base = (size_t)node * DIM + lane * 4;

  float4 t0 = *(const float4*)&p0[base];
  float4 t1 = *(const float4*)&p1[base];
  float4 t;
  t.x = t0.x + scale1 * t1.x; t.y = t0.y + scale1 * t1.y;
  t.z = t0.z + scale1 * t1.z; t.w = t0.w + scale1 * t1.w;
  if (p2) {
    float4 t2 = *(const float4*)&p2[base];
    t.x += t2.x; t.y += t2.y; t.z += t2.z; t.w += t2.w;
  }
  float s  = t.x + t.y + t.z + t.w;
  float ss = t.x * t.x + t.y * t.y + t.z * t.z + t.w * t.w;
  s  = wave_sum32(s);
  ss = wave_sum32(ss);
  const float mean = s * (1.0f / DIM);
  const float var  = ss * (1.0f / DIM) - mean * mean;
  const float rstd = rsqrtf(var + 1e-5f);

  const float4 g4 = *(const float4*)&g[lane * 4];
  const float4 b4 = *(const float4*)&b[lane * 4];
  float4 o;
  o.x = (t.x - mean) * rstd * g4.x + b4.x;
  o.y = (t.y - mean) * rstd * g4.y + b4.y;
  o.z = (t.z - mean) * rstd * g4.z + b4.z;
  o.w = (t.w - mean) * rstd * g4.w + b4.w;
  *(float4*)&out[base] = o;
}

// ---------------------------------------------------------------------------
extern "C" void kernel_launch(void* const* d_in, const int* in_sizes, int n_in,
                              void* d_out, int out_size, void* d_ws, size_t ws_size,
                              hipStream_t stream) {
  (void)in_sizes; (void)n_in; (void)out_size; (void)ws_size;
  const float*     x     = (const float*)d_in[0];
  const long long* ei    = (const long long*)d_in[1];   // int64 edge_index [2,E]
  const float*     Wq    = (const float*)d_in[2];
  const float*     bq    = (const float*)d_in[3];
  const float*     Wk    = (const float*)d_in[4];
  const float*     bk    = (const float*)d_in[5];
  const float*     Wv    = (const float*)d_in[6];
  const float*     bvv   = (const float*)d_in[7];
  const float*     Wskip = (const float*)d_in[8];
  const float*     bskip = (const float*)d_in[9];
  const float*     g1    = (const float*)d_in[10];
  const float*     beta1 = (const float*)d_in[11];
  const float*     W1    = (const float*)d_in[12];
  const float*     b1    = (const float*)d_in[13];
  const float*     W2    = (const float*)d_in[14];
  const float*     b2    = (const float*)d_in[15];
  const float*     g2    = (const float*)d_in[16];
  const float*     beta2 = (const float*)d_in[17];
  float* out = (float*)d_out;

  // Workspace carving
  float* ws = (float*)d_ws;
  size_t off = 0;
  float* q    = ws + off; off += (size_t)N_NODES * HEADS * CH;
  float* kbuf = ws + off; off += (size_t)N_NODES * HEADS * CH;
  float* vbuf = ws + off; off += (size_t)N_NODES * HEADS * CH;
  float* skip = ws + off; off += (size_t)N_NODES * DIM;
  float* hbuf = ws + off; off += (size_t)N_NODES * DIM;
  float* alpha= ws + off; off += (size_t)N_EDGES * HEADS;
  unsigned int* m_enc = (unsigned int*)(ws + off); off += (size_t)N_NODES * HEADS;
  float* denom= ws + off; off += (size_t)N_NODES * HEADS;
  float* agg  = ws + off; off += (size_t)N_NODES * DIM;
  float* ff1  = ws + off; off += (size_t)N_NODES * FF_DIM;
  float* ff2  = ws + off; off += (size_t)N_NODES * DIM;

  // Re-init accumulation buffers every call (graph-capture safe)
  (void)hipMemsetAsync(m_enc, 0, (size_t)N_NODES * HEADS * 4, stream);
  (void)hipMemsetAsync(denom, 0, (size_t)N_NODES * HEADS * 4, stream);
  (void)hipMemsetAsync(agg,   0, (size_t)N_NODES * DIM   * 4, stream);

  const int MT = N_NODES / 16;                        // 625 (exact)

  // QKV + skip projections (fp32 WMMA, fully unrolled)
  gemm_wmma_f32<DIM, HEADS * CH, 0><<<dim3(MT, (HEADS * CH) / 128), 256, 0, stream>>>(x, Wq, bq, q);
  gemm_wmma_f32<DIM, HEADS * CH, 0><<<dim3(MT, (HEADS * CH) / 128), 256, 0, stream>>>(x, Wk, bk, kbuf);
  gemm_wmma_f32<DIM, HEADS * CH, 0><<<dim3(MT, (HEADS * CH) / 128), 256, 0, stream>>>(x, Wv, bvv, vbuf);
  gemm_wmma_f32<DIM, DIM, 0><<<dim3(MT, 1), 256, 0, stream>>>(x, Wskip, bskip, skip);

  // Attention: logits+max, exp+denom, weighted aggregation
  edge_logits_max<<<(N_EDGES * HEADS) / 8, 256, 0, stream>>>(q, kbuf, ei, alpha, m_enc);
  edge_exp_denom<<<((size_t)N_EDGES * HEADS + 255) / 256, 256, 0, stream>>>(ei, alpha, m_enc, denom);
  edge_aggregate<<<N_EDGES / 8, 256, 0, stream>>>(ei, alpha, denom, vbuf, agg);

  // h = LN(x + agg/H + skip)
  ln_fused<<<N_NODES / 8, 256, 0, stream>>>(x, agg, 1.0f / HEADS, skip, g1, beta1, hbuf);

  // FFN
  gemm_wmma_f32<DIM, FF_DIM, 1><<<dim3(MT, FF_DIM / 128), 256, 0, stream>>>(hbuf, W1, b1, ff1);
  gemm_wmma_f32<FF_DIM, DIM, 0><<<dim3(MT, 1), 256, 0, stream>>>(ff1, W2, b2, ff2);

  // out = LN(h + ff)
  ln_fused<<<N_NODES / 8, 256, 0, stream>>>(hbuf, ff2, 1.0f, nullptr, g2, beta2, out);
}